// VisionTransformer_19198503813211
// MI455X (gfx1250) — compile-verified
//
#include <hip/hip_runtime.h>

typedef float v2f __attribute__((ext_vector_type(2)));
typedef float v8f __attribute__((ext_vector_type(8)));

#define NB      256           // batch
#define SEQ     65
#define HID     512
#define HEADS   8
#define ROWS    (NB*SEQ)      // 16640
#define LNELEM  (SEQ*HID)     // 33280 elements per sample
#define SCLD    80            // padded scores row stride (65 -> 80)
#define SC_SMP  (SEQ*SCLD)    // 5200 per-sample scores stride
#define KC      32            // K-chunk for LDS-tiled GEMM
#define BPAD    68            // LDS row pad (68*4 bytes, keeps 16B alignment)

__device__ __forceinline__ v8f wmma_acc(v8f c, v2f a, v2f b) {
    // v_wmma_f32_16x16x4_f32 (fp32 A/B, fp32 accum) — full fp32 precision
    return __builtin_amdgcn_wmma_f32_16x16x4_f32(false, a, false, b, (short)0, c,
                                                 false, false);
}

// Issue the async global->LDS copies for one B chunk: B[k0..k0+31][nBase..+63]
// 32x64 floats = 512 b128 slots; 128 threads x 4 slots each. ASYNCcnt-tracked.
__device__ __forceinline__ void stage_chunk(const float* __restrict__ B,
                                            int k0, int nBase, int tid,
                                            float (*buf)[BPAD])
{
    #pragma unroll
    for (int q = 0; q < 4; ++q) {
        const int slot = tid * 4 + q;          // 0..511
        const int r    = slot >> 4;            // 0..31  (chunk row)
        const int c4   = slot & 15;            // 0..15  (x4 floats)
        const unsigned long long ga =
            (unsigned long long)(B + (size_t)(k0 + r) * HID + nBase + c4 * 4);
        const unsigned int loff =
            (unsigned int)(size_t)&buf[r][c4 * 4];   // low 32 bits = ds offset
        asm volatile("global_load_async_to_lds_b128 %0, %1, off"
                     :: "v"(loff), "v"(ga) : "memory");
    }
}

// ---------------------------------------------------------------------------
// LDS-tiled WMMA GEMM for the dominant shapes: D(Mx512) = A(Mx512)*B(512x512)
// [+bias] [+=D].  Block = 128 threads = 4 waves.  Each wave owns one 16-row
// m-tile; the block shares a 64-column strip of B.  B chunks are staged with
// CDNA5 async global->LDS copies, DOUBLE-BUFFERED so chunk i+1's DMA overlaps
// chunk i's 32 v_wmma_f32_16x16x4_f32 ops (A frag reused 4x per k-step, each
// LDS B element reused by 4 waves).  Requires M % 64 == 0 (ROWS = 260*64).
// ---------------------------------------------------------------------------
__global__ void __launch_bounds__(128)
k_gemm_big(const float* __restrict__ A, const float* __restrict__ B,
           float* __restrict__ D, const float* __restrict__ bias, int accum)
{
    __shared__ __attribute__((aligned(16))) float sB[2][KC][BPAD];

    const int tid  = threadIdx.x;
    const int wave = tid >> 5;
    const int lane = tid & 31;
    const int half = lane >> 4;
    const int pos  = lane & 15;
    const int m0   = (blockIdx.x * 4 + wave) * 16;
    const int nBase = blockIdx.y * 64;
    const float* Arow = A + (size_t)(m0 + pos) * HID;

    const v8f vz = {0.f,0.f,0.f,0.f,0.f,0.f,0.f,0.f};
    v8f c0 = vz, c1 = vz, c2 = vz, c3 = vz;

    // Prologue: stage chunk 0, wait, publish.
    stage_chunk(B, 0, nBase, tid, sB[0]);
    asm volatile("s_wait_asynccnt 0" ::: "memory");
    __syncthreads();

    for (int k0 = 0; k0 < HID; k0 += KC) {
        const int cur = (k0 / KC) & 1;
        // Fire-and-forget DMA of the NEXT chunk into the other buffer
        // (safe: everyone finished reading it at the previous barrier).
        if (k0 + KC < HID)
            stage_chunk(B, k0 + KC, nBase, tid, sB[cur ^ 1]);

        // Compute current chunk: 8 k-steps x 4 column sub-tiles.
        #pragma unroll
        for (int ks = 0; ks < KC; ks += 4) {
            const int ka = k0 + ks + 2 * half;
            v2f a;
            a.x = Arow[ka];
            a.y = Arow[ka + 1];
            const int kl = ks + 2 * half;
            v2f b0, b1, b2, b3;
            b0.x = sB[cur][kl][pos];        b0.y = sB[cur][kl + 1][pos];
            b1.x = sB[cur][kl][16 + pos];   b1.y = sB[cur][kl + 1][16 + pos];
            b2.x = sB[cur][kl][32 + pos];   b2.y = sB[cur][kl + 1][32 + pos];
            b3.x = sB[cur][kl][48 + pos];   b3.y = sB[cur][kl + 1][48 + pos];
            c0 = wmma_acc(c0, a, b0);
            c1 = wmma_acc(c1, a, b1);
            c2 = wmma_acc(c2, a, b2);
            c3 = wmma_acc(c3, a, b3);
        }

        // Next chunk's DMA complete + all waves done with current chunk.
        asm volatile("s_wait_asynccnt 0" ::: "memory");
        __syncthreads();
    }

    // --- epilogue: 4 column tiles, all rows/cols in range by construction ---
    v8f acc[4] = {c0, c1, c2, c3};
    #pragma unroll
    for (int t = 0; t < 4; ++t) {
        const int col = nBase + t * 16 + pos;
        const float bb = bias ? bias[col] : 0.f;
        #pragma unroll
        for (int i = 0; i < 8; ++i) {
            const int row = m0 + i + 8 * half;     // VGPR i -> M=i / M=i+8
            float v = acc[t][i] + bb;
            if (accum) v += D[(size_t)row * HID + col];
            D[(size_t)row * HID + col] = v;
        }
    }
}

// ---------------------------------------------------------------------------
// Generic batched WMMA-f32 GEMM (used for the small attention GEMMs):
// D = alpha * A(MxK) * B(KxN);  BT=true: B stored (N x K) row-major.
// ---------------------------------------------------------------------------
template<bool BT>
__global__ void __launch_bounds__(128)
k_gemm(const float* __restrict__ A, int lda, int sA,
       const float* __restrict__ B, int ldb, int sB,
       float* __restrict__ D, int ldd, int sD,
       int M, int N, int K, float alpha)
{
    const int wave = threadIdx.x >> 5;
    const int lane = threadIdx.x & 31;
    const int half = lane >> 4;
    const int pos  = lane & 15;
    const int mt   = blockIdx.x * 4 + wave;
    if (mt * 16 >= M) return;                    // wave-uniform exit
    const int z = blockIdx.z;
    A += (size_t)z * (size_t)sA;
    B += (size_t)z * (size_t)sB;
    D += (size_t)z * (size_t)sD;
    const int m0 = mt * 16, n0 = blockIdx.y * 16;
    const int arow = m0 + pos;
    const int bcol = n0 + pos;

    v8f c = {0.f,0.f,0.f,0.f,0.f,0.f,0.f,0.f};
    for (int kk = 0; kk < K; kk += 4) {
        const int ka = kk + 2 * half;
        v2f a, b;
        a.x = (arow < M && ka     < K) ? A[(size_t)arow * lda + ka]     : 0.f;
        a.y = (arow < M && ka + 1 < K) ? A[(size_t)arow * lda + ka + 1] : 0.f;
        if (BT) {
            b.x = (bcol < N && ka     < K) ? B[(size_t)bcol * ldb + ka]     : 0.f;
            b.y = (bcol < N && ka + 1 < K) ? B[(size_t)bcol * ldb + ka + 1] : 0.f;
        } else {
            b.x = (bcol < N && ka     < K) ? B[(size_t)(ka)     * ldb + bcol] : 0.f;
            b.y = (bcol < N && ka + 1 < K) ? B[(size_t)(ka + 1) * ldb + bcol] : 0.f;
        }
        c = wmma_acc(c, a, b);
    }

    if (bcol < N) {
        #pragma unroll
        for (int i = 0; i < 8; ++i) {
            const int row = m0 + i + 8 * half;
            if (row < M)
                D[(size_t)row * ldd + bcol] = c[i] * alpha;
        }
    }
}

// ---------------------------------------------------------------------------
// Patch embedding: Z[n,1+p,:] = X_patches[n,p,:]@Wp + bp + pos[1+p,:]
// X flat is exactly [16384 x 256] row-major. Grid (256, 32), block 128.
// ---------------------------------------------------------------------------
__global__ void __launch_bounds__(128)
k_patch_embed(const float* __restrict__ X, const float* __restrict__ Wp,
              const float* __restrict__ bp, const float* __restrict__ pos,
              float* __restrict__ Z)
{
    const int wave = threadIdx.x >> 5;
    const int lane = threadIdx.x & 31;
    const int half = lane >> 4;
    const int p16  = lane & 15;
    const int m0   = (blockIdx.x * 4 + wave) * 16;
    const int n0   = blockIdx.y * 16;
    const int arow = m0 + p16;
    const int bcol = n0 + p16;

    v8f c = {0.f,0.f,0.f,0.f,0.f,0.f,0.f,0.f};
    for (int kk = 0; kk < 256; kk += 4) {
        const int ka = kk + 2 * half;
        v2f a, b;
        a.x = X[(size_t)arow * 256 + ka];
        a.y = X[(size_t)arow * 256 + ka + 1];
        b.x = Wp[(size_t)(ka)     * HID + bcol];
        b.y = Wp[(size_t)(ka + 1) * HID + bcol];
        c = wmma_acc(c, a, b);
    }
    #pragma unroll
    for (int i = 0; i < 8; ++i) {
        const int row = m0 + i + 8 * half;
        const int n = row >> 6, p = row & 63;
        Z[((size_t)n * SEQ + 1 + p) * HID + bcol] =
            c[i] + bp[bcol] + pos[(size_t)(1 + p) * HID + bcol];
    }
}

// Z[n,0,:] = cls + pos[0,:]
__global__ void k_cls_init(float* __restrict__ Z, const float* __restrict__ cls,
                           const float* __restrict__ pos)
{
    const int n = blockIdx.x;
    for (int c = threadIdx.x; c < HID; c += blockDim.x)
        Z[(size_t)n * LNELEM + c] = cls[c] + pos[c];
}

// LayerNorm over (seq, hidden) per sample, affine (SEQ, HID). One block/sample.
__global__ void k_ln2d(const float* __restrict__ src, float* __restrict__ dst,
                       const float* __restrict__ w, const float* __restrict__ b)
{
    __shared__ float sh1[256];
    __shared__ float sh2[256];
    const int n   = blockIdx.x;
    const int tid = threadIdx.x;
    const float* x = src + (size_t)n * LNELEM;
    float s = 0.f, s2 = 0.f;
    for (int i = tid; i < LNELEM; i += 256) {
        const float v = x[i];
        s += v; s2 += v * v;
    }
    sh1[tid] = s; sh2[tid] = s2;
    __syncthreads();
    for (int off = 128; off > 0; off >>= 1) {
        if (tid < off) { sh1[tid] += sh1[tid + off]; sh2[tid] += sh2[tid + off]; }
        __syncthreads();
    }
    const float mu   = sh1[0] * (1.f / (float)LNELEM);
    const float var  = sh2[0] * (1.f / (float)LNELEM) - mu * mu;
    const float rstd = rsqrtf(var + 1e-5f);
    float* y = dst + (size_t)n * LNELEM;
    for (int i = tid; i < LNELEM; i += 256)
        y[i] = (x[i] - mu) * rstd * w[i] + b[i];
}

// acc = Z + bo (residual folded in before per-head Wo accumulation)
__global__ void k_res_init(float* __restrict__ acc, const float* __restrict__ Z,
                           const float* __restrict__ bo)
{
    const size_t total = (size_t)ROWS * HID;
    for (size_t i = (size_t)blockIdx.x * blockDim.x + threadIdx.x; i < total;
         i += (size_t)gridDim.x * blockDim.x)
        acc[i] = Z[i] + bo[i & (HID - 1)];
}

// Softmax over 65 logits per row (stride 80). One wave32 per row, block = 4 rows.
__global__ void k_softmax(float* __restrict__ sc)
{
    const int row = blockIdx.x * 4 + (threadIdx.x >> 5);
    if (row >= ROWS) return;
    const int lane = threadIdx.x & 31;
    float* p = sc + (size_t)row * SCLD;
    const float a0 = p[lane];
    const float a1 = (lane + 32 < SEQ) ? p[lane + 32] : -3.4e38f;
    const float a2 = (lane + 64 < SEQ) ? p[lane + 64] : -3.4e38f;
    float m = fmaxf(a0, fmaxf(a1, a2));
    for (int off = 16; off > 0; off >>= 1) m = fmaxf(m, __shfl_xor(m, off, 32));
    const float e0 = expf(a0 - m);
    const float e1 = (lane + 32 < SEQ) ? expf(a1 - m) : 0.f;
    const float e2 = (lane + 64 < SEQ) ? expf(a2 - m) : 0.f;
    float s = e0 + e1 + e2;
    for (int off = 16; off > 0; off >>= 1) s += __shfl_xor(s, off, 32);
    const float inv = 1.f / s;
    p[lane] = e0 * inv;
    if (lane + 32 < SEQ) p[lane + 32] = e1 * inv;
    if (lane + 64 < SEQ) p[lane + 64] = e2 * inv;
}

// out[n,:] = tanh(Z[n,0,:] @ Wh + bh)  (512x10, tiny)
__global__ void k_head(const float* __restrict__ Z, const float* __restrict__ Wh,
                       const float* __restrict__ bh, float* __restrict__ out)
{
    const int n = blockIdx.x;
    const int j = threadIdx.x;
    if (j < 10) {
        const float* z = Z + (size_t)n * LNELEM;   // seq position 0
        float s = bh[j];
        for (int c = 0; c < HID; ++c) s += z[c] * Wh[c * 10 + j];
        out[n * 10 + j] = tanhf(s);
    }
}

extern "C" void kernel_launch(void* const* d_in, const int* in_sizes, int n_in,
                              void* d_out, int out_size, void* d_ws, size_t ws_size,
                              hipStream_t stream)
{
    const float* X     = (const float*)d_in[0];
    const float* Wp    = (const float*)d_in[1];
    const float* bp    = (const float*)d_in[2];
    const float* cls   = (const float*)d_in[3];
    const float* pos   = (const float*)d_in[4];
    const float* ln1_w = (const float*)d_in[5];
    const float* ln1_b = (const float*)d_in[6];
    const float* Wq    = (const float*)d_in[7];
    const float* bq    = (const float*)d_in[8];
    const float* Wk    = (const float*)d_in[9];
    const float* bk    = (const float*)d_in[10];
    const float* Wv    = (const float*)d_in[11];
    const float* bv    = (const float*)d_in[12];
    const float* Wo    = (const float*)d_in[13];
    const float* bo    = (const float*)d_in[14];
    const float* ln2_w = (const float*)d_in[15];
    const float* ln2_b = (const float*)d_in[16];
    const float* W2    = (const float*)d_in[17];
    const float* b2    = (const float*)d_in[18];
    const float* Wh    = (const float*)d_in[19];
    const float* bh    = (const float*)d_in[20];
    float* out = (float*)d_out;

    const size_t ZSZ = (size_t)NB * LNELEM;   // 8,519,680 floats
    float* ws  = (float*)d_ws;
    float* Z   = ws;
    float* Zn  = Z   + ZSZ;
    float* Q   = Zn  + ZSZ;                   // also reused for attn@v output
    float* Kb  = Q   + ZSZ;
    float* Vb  = Kb  + ZSZ;
    float* ACC = Vb  + ZSZ;
    float* SC  = ACC + ZSZ;                   // NB * 65 * 80

    const float scale = 1.f / (512.f * 512.f);
    const dim3 blk(128);
    const dim3 gBig(ROWS / 16 / 4, HID / 64, 1);    // 260 x 8, LDS-tiled GEMM
    const dim3 gScr(2, 5, NB);                      // scores: 5x5 tiles of 65
    const dim3 gAV (2, HID / 16, NB);               // attn @ v

    // --- Embedding ---
    k_cls_init<<<NB, 256, 0, stream>>>(Z, cls, pos);
    k_patch_embed<<<dim3((NB * 64) / 16 / 4, HID / 16), blk, 0, stream>>>(X, Wp, bp, pos, Z);

    // --- Transformer blocks (shared weights) ---
    for (int b = 0; b < 6; ++b) {
        k_ln2d<<<NB, 256, 0, stream>>>(Z, Zn, ln1_w, ln1_b);
        k_res_init<<<2048, 256, 0, stream>>>(ACC, Z, bo);
        for (int h = 0; h < HEADS; ++h) {
            const float* Wq_h = Wq + (size_t)h * HID * HID;
            const float* Wk_h = Wk + (size_t)h * HID * HID;
            const float* Wv_h = Wv + (size_t)h * HID * HID;
            const float* Wo_h = Wo + (size_t)h * HID * HID;  // rows h*512..h*512+511
            // q/k/v = Zn @ W + bias  (double-buffered async LDS GEMM)
            k_gemm_big<<<gBig, blk, 0, stream>>>(Zn, Wq_h, Q,  bq + h * HID, 0);
            k_gemm_big<<<gBig, blk, 0, stream>>>(Zn, Wk_h, Kb, bk + h * HID, 0);
            k_gemm_big<<<gBig, blk, 0, stream>>>(Zn, Wv_h, Vb, bv + h * HID, 0);
            // scores = scale * q @ k^T   (per-sample batched, B transposed)
            k_gemm<true><<<gScr, blk, 0, stream>>>(Q, HID, LNELEM, Kb, HID, LNELEM,
                                                   SC, SCLD, SC_SMP,
                                                   SEQ, SEQ, HID, scale);
            k_softmax<<<ROWS / 4, 128, 0, stream>>>(SC);
            // out = attn @ v  (overwrites Q, which is no longer needed)
            k_gemm<false><<<gAV, blk, 0, stream>>>(SC, SCLD, SC_SMP, Vb, HID, LNELEM,
                                                   Q, HID, LNELEM,
                                                   SEQ, HID, SEQ, 1.f);
            // acc += out @ Wo_h
            k_gemm_big<<<gBig, blk, 0, stream>>>(Q, Wo_h, ACC, nullptr, 1);
        }
        k_ln2d<<<NB, 256, 0, stream>>>(ACC, Zn, ln2_w, ln2_b);
        // Z = LN2(res1 + Z) @ W2 + b2   (no residual after, per reference)
        k_gemm_big<<<gBig, blk, 0, stream>>>(Zn, W2, Z, b2, 0);
    }

    // --- Classification head ---
    k_head<<<NB, 32, 0, stream>>>(Z, Wh, bh, out);
}